// TopKTopPSampler_2456721293541
// MI455X (gfx1250) — compile-verified
//
#include <hip/hip_runtime.h>
#include <math.h>

#define BLOCK   1024
#define NVOCAB  128000
#define N4      (NVOCAB / 4)          // 32000 float4 per row
#define NTILES  (N4 / BLOCK)          // 31 full async tiles
#define NREM    (N4 - NTILES * BLOCK) // 256 remainder float4
#define NB1     4096                  // coarse histogram bins (key[31:20])
#define CAP     4096                  // candidate capacity (>= k_max + max bucket count)

typedef float v4f __attribute__((ext_vector_type(4)));

// ---- monotone float <-> sortable-uint key mapping (unsigned ascending == float ascending)
__device__ __forceinline__ unsigned f2k(float f) {
    unsigned u = __float_as_uint(f);
    return u ^ (unsigned)(((int)u >> 31) | (int)0x80000000);
}
__device__ __forceinline__ float k2f(unsigned k) {
    unsigned m = (unsigned)(~((int)k >> 31)) | 0x80000000u;
    return __uint_as_float(k ^ m);
}

struct StageBufs { v4f stage[2][BLOCK]; };                    // 32 KB (pass A only)
struct CandBufs  { unsigned cand[CAP]; float w[CAP]; };       // 32 KB (pass B+)
union ShUnion { StageBufs a; CandBufs c; };

// Inclusive suffix scan of h[0..nb) in place, then find bucket b = max j with
// suffix(j) >= kneed. All threads participate.
__device__ void suffix_select(unsigned* h, int nb, unsigned kneed,
                              unsigned* outBucket, int tid) {
    __syncthreads();
    for (int off = 1; off < nb; off <<= 1) {
        unsigned tmp[NB1 / BLOCK];
        int c = 0;
        for (int j = tid; j < nb; j += BLOCK, ++c)
            tmp[c] = h[j] + ((j + off < nb) ? h[j + off] : 0u);
        __syncthreads();
        c = 0;
        for (int j = tid; j < nb; j += BLOCK, ++c) h[j] = tmp[c];
        __syncthreads();
    }
    for (int j = tid; j < nb; j += BLOCK) {
        unsigned nxt = (j + 1 < nb) ? h[j + 1] : 0u;
        if (h[j] >= kneed && nxt < kneed) *outBucket = (unsigned)j;
    }
    __syncthreads();
}

__global__ __launch_bounds__(BLOCK)
void topk_topp_radix_kernel(const float* __restrict__ logits,
                            const int*   __restrict__ kArr,
                            const float* __restrict__ pArr,
                            float*       __restrict__ out) {
    __shared__ ShUnion sh;
    __shared__ unsigned hist[NB1];
    __shared__ unsigned sMaxKey, sCount, sBucket, sM, sKthKey, sThreshKey;

    const int tid = threadIdx.x;
    const int row = blockIdx.x;
    const float* rowPtr = logits + (size_t)row * NVOCAB;
    const v4f* row4     = (const v4f*)rowPtr;
    v4f* out4           = (v4f*)(out + (size_t)row * NVOCAB);

    int kk = kArr[row];
    kk = kk < 1 ? 1 : (kk > NVOCAB ? NVOCAB : kk);
    const float pv = pArr[row];

    for (int j = tid; j < NB1; j += BLOCK) hist[j] = 0u;
    if (tid == 0) { sMaxKey = 0u; sCount = 0u; }
    __syncthreads();

    // -------- Pass A: async-staged stream -> row max + histogram of key[31:20] --------
    unsigned ldsA[2];
    ldsA[0] = (unsigned)(size_t)&sh.a.stage[0][tid];
    ldsA[1] = (unsigned)(size_t)&sh.a.stage[1][tid];

#define ISSUE_ASYNC(tile, b) do {                                              \
        unsigned goff_ = (unsigned)(((tile) * BLOCK + tid) * 16);              \
        asm volatile("global_load_async_to_lds_b128 %0, %1, %2"                \
                     :: "v"(ldsA[(b)]), "v"(goff_), "s"(rowPtr) : "memory");   \
    } while (0)

    unsigned lmax = 0u;
    {
        int buf = 0;
        ISSUE_ASYNC(0, 0);
        for (int t = 0; t < NTILES; ++t) {
            if (t + 1 < NTILES) {
                ISSUE_ASYNC(t + 1, buf ^ 1);
                asm volatile("s_wait_asynccnt 1" ::: "memory");
            } else {
                asm volatile("s_wait_asynccnt 0" ::: "memory");
            }
            v4f v = sh.a.stage[buf][tid];
            unsigned k0 = f2k(v.x), k1 = f2k(v.y), k2 = f2k(v.z), k3 = f2k(v.w);
            lmax = max(max(lmax, k0), max(k1, max(k2, k3)));
            atomicAdd(&hist[k0 >> 20], 1u);
            atomicAdd(&hist[k1 >> 20], 1u);
            atomicAdd(&hist[k2 >> 20], 1u);
            atomicAdd(&hist[k3 >> 20], 1u);
            buf ^= 1;
        }
        if (tid < NREM) {
            v4f v = row4[NTILES * BLOCK + tid];
            unsigned k0 = f2k(v.x), k1 = f2k(v.y), k2 = f2k(v.z), k3 = f2k(v.w);
            lmax = max(max(lmax, k0), max(k1, max(k2, k3)));
            atomicAdd(&hist[k0 >> 20], 1u);
            atomicAdd(&hist[k1 >> 20], 1u);
            atomicAdd(&hist[k2 >> 20], 1u);
            atomicAdd(&hist[k3 >> 20], 1u);
        }
    }
    atomicMax(&sMaxKey, lmax);
    __syncthreads();
    const float M = k2f(sMaxKey);

    // coarse bucket holding the k-th largest element
    suffix_select(hist, NB1, (unsigned)kk, &sBucket, tid);
    const unsigned b1 = sBucket;

    // -------- Pass B (L2-resident): collect ALL keys with top-12 prefix >= b1 --------
    for (int j = tid; j < CAP; j += BLOCK) sh.c.cand[j] = 0u;
    __syncthreads();
    for (int i4 = tid; i4 < N4; i4 += BLOCK) {
        v4f v = row4[i4];
        unsigned ks[4] = { f2k(v.x), f2k(v.y), f2k(v.z), f2k(v.w) };
#pragma unroll
        for (int e = 0; e < 4; ++e) {
            if ((ks[e] >> 20) >= b1) {
                unsigned pos = atomicAdd(&sCount, 1u);
                if (pos < CAP) sh.c.cand[pos] = ks[e];
            }
        }
    }
    __syncthreads();

    // -------- Bitonic sort candidates descending (padded with key 0) --------
    for (unsigned size = 2; size <= CAP; size <<= 1) {
        for (unsigned stride = size >> 1; stride > 0; stride >>= 1) {
            __syncthreads();
            for (unsigned i = tid; i < CAP; i += BLOCK) {
                unsigned p = i ^ stride;
                if (p > i) {
                    bool desc = ((i & size) == 0);
                    unsigned a = sh.c.cand[i], b = sh.c.cand[p];
                    if (desc ? (a < b) : (a > b)) { sh.c.cand[i] = b; sh.c.cand[p] = a; }
                }
            }
        }
    }
    __syncthreads();

    // k-th largest value's key, directly from the sorted candidates
    if (tid == 0) sKthKey = sh.c.cand[kk - 1];
    __syncthreads();
    const unsigned kthKey = sKthKey;

    // -------- exp weights over survivors + inclusive prefix scan (Hillis-Steele) --------
    for (int i = tid; i < CAP; i += BLOCK) {
        unsigned key = sh.c.cand[i];
        sh.c.w[i] = (key >= kthKey) ? expf(k2f(key) - M) : 0.0f;
    }
    __syncthreads();
    for (int off = 1; off < CAP; off <<= 1) {
        float tmp[CAP / BLOCK];
        int c = 0;
        for (int i = tid; i < CAP; i += BLOCK, ++c)
            tmp[c] = sh.c.w[i] + ((i >= off) ? sh.c.w[i - off] : 0.0f);
        __syncthreads();
        c = 0;
        for (int i = tid; i < CAP; i += BLOCK, ++c) sh.c.w[i] = tmp[c];
        __syncthreads();
    }
    const float Z  = sh.c.w[CAP - 1];
    const float pZ = pv * Z;
    if (tid == 0) sM = 0u;
    __syncthreads();
    // keep element i (descending order) iff survivor and mass strictly above it < p*Z
    for (int i = tid; i < CAP; i += BLOCK) {
        unsigned key = sh.c.cand[i];
        float excl = (i > 0) ? sh.c.w[i - 1] : 0.0f;
        if (key >= kthKey && excl < pZ) atomicAdd(&sM, 1u);
    }
    __syncthreads();
    if (tid == 0) sThreshKey = sh.c.cand[sM - 1];   // smallest kept value's key
    __syncthreads();
    const unsigned thK = sThreshKey;

    // -------- Pass E: write masked logits (non-temporal: last use of input,
    //          and don't let the 131MB output evict other rows from L2) --------
    const float NEG_INF = -__builtin_inff();
    for (int i4 = tid; i4 < N4; i4 += BLOCK) {
        v4f v = __builtin_nontemporal_load(&row4[i4]);
        v4f o;
        o.x = (f2k(v.x) >= thK) ? v.x : NEG_INF;
        o.y = (f2k(v.y) >= thK) ? v.y : NEG_INF;
        o.z = (f2k(v.z) >= thK) ? v.z : NEG_INF;
        o.w = (f2k(v.w) >= thK) ? v.w : NEG_INF;
        __builtin_nontemporal_store(o, &out4[i4]);
    }
#undef ISSUE_ASYNC
}

extern "C" void kernel_launch(void* const* d_in, const int* in_sizes, int n_in,
                              void* d_out, int out_size, void* d_ws, size_t ws_size,
                              hipStream_t stream) {
    const float* logits = (const float*)d_in[0];
    const int*   k      = (const int*)d_in[1];
    const float* p      = (const float*)d_in[2];
    float*       out    = (float*)d_out;
    const int B = in_sizes[1];   // 256 rows
    hipLaunchKernelGGL(topk_topp_radix_kernel, dim3(B), dim3(BLOCK), 0, stream,
                       logits, k, p, out);
}